// DiffPoolNet_28707561406850
// MI455X (gfx1250) — compile-verified
//
#include <hip/hip_runtime.h>
#include <hip/hip_bf16.h>

typedef __attribute__((ext_vector_type(2))) float v2f;
typedef __attribute__((ext_vector_type(8))) float v8f;

// ---------------------------------------------------------------- elementwise
__global__ void fill_f32(float* p, float v, long long n) {
  long long i = (long long)blockIdx.x * blockDim.x + threadIdx.x;
  if (i < n) p[i] = v;
}

__global__ void deg_count(float* __restrict__ deg, const int* __restrict__ col, int n) {
  int i = blockIdx.x * blockDim.x + threadIdx.x;
  if (i < n) atomicAdd(&deg[col[i]], 1.0f);
}

__global__ void deg_to_dinv(float* p, int n) {
  int i = blockIdx.x * blockDim.x + threadIdx.x;
  if (i < n) {
    float d = p[i];
    p[i] = d > 0.0f ? rsqrtf(d) : 0.0f;
  }
}

// Y[i][:] = dinv[i]^2 * X[i][:]   (self-loop term of symmetric-normalized prop)
__global__ void prop_selfinit(const float* __restrict__ X, float* __restrict__ Y,
                              const float* __restrict__ dinv, long long n64) {
  long long i = (long long)blockIdx.x * blockDim.x + threadIdx.x;
  if (i < n64) {
    float di = dinv[i >> 6];
    Y[i] = di * di * X[i];
  }
}

__global__ void bias_act(float* __restrict__ Y, const float* __restrict__ b,
                         long long n64, int do_relu) {
  long long i = (long long)blockIdx.x * blockDim.x + threadIdx.x;
  if (i < n64) {
    float v = Y[i] + b[i & 63];
    Y[i] = do_relu ? fmaxf(v, 0.0f) : v;
  }
}

// ---------------------------------------------------------------- edge scatter
// One wave per edge: Y[didx[e]][:] += w * X[sidx[e]][:]   (64 floats, float2/lane)
__global__ void edge_scatter(const float* __restrict__ X, float* __restrict__ Y,
                             const int* __restrict__ sidx, const int* __restrict__ didx,
                             const float* __restrict__ dinv, int nedges, int use_norm) {
  long long t = (long long)blockIdx.x * blockDim.x + threadIdx.x;
  int e = (int)(t >> 5);
  if (e >= nedges) return;            // uniform per wave
  int lane = (int)t & 31;
  int s = sidx[e];
  int d = didx[e];
  float w = use_norm ? (dinv[s] * dinv[d]) : 1.0f;
  float2 v = ((const float2*)(X + (size_t)s * 64))[lane];
  float* y = Y + (size_t)d * 64 + lane * 2;
  atomicAdd(y + 0, w * v.x);
  atomicAdd(y + 1, w * v.y);
}

// ---------------------------------------------------------------- WMMA GEMM
// Y[nrows,64] = X[nrows,64] @ W[64,64], fp32 via V_WMMA_F32_16X16X4_F32.
// 4 waves/block; each wave owns a 16-row strip and the full 64-wide output.
// Weight staged in LDS k-pair-interleaved so each B fragment {W[k][n],W[k+1][n]}
// is one aligned ds_load_b64 straight into an even VGPR pair.
__global__ __launch_bounds__(128) void gemm64(const float* __restrict__ X,
                                              const float* __restrict__ W,
                                              float* __restrict__ Y, int nrows) {
  __shared__ float sW2[64 * 64];      // sW2[(k>>1)*128 + n*2 + (k&1)] = W[k][n]
  int tid = threadIdx.x;
  for (int i = tid; i < 64 * 64; i += 128) {
    int k = i >> 6;
    int n = i & 63;
    sW2[((k >> 1) << 7) + (n << 1) + (k & 1)] = W[i];
  }
  __syncthreads();

  int wave = tid >> 5;
  int lane = tid & 31;
  int r0 = (blockIdx.x * 4 + wave) * 16;
  if (r0 >= nrows) return;            // uniform per wave; EXEC stays all-ones

  int m    = lane & 15;               // A: M index
  int koff = (lane >> 4) << 1;        // A: lanes 0-15 -> K 0,1 ; lanes 16-31 -> K 2,3
  int nc   = lane & 15;               // B/C: N index

  v8f acc0 = {}, acc1 = {}, acc2 = {}, acc3 = {};
  const float* xrow = X + (size_t)(r0 + m) * 64;

  for (int kb = 0; kb < 64; kb += 4) {
    v2f a = *(const v2f*)(xrow + kb + koff);                 // one global_load_b64
    const v2f* wrow = (const v2f*)(sW2 + (((kb + koff) >> 1) << 7));
    v2f b0 = wrow[nc + 0];                                   // one ds_load_b64 each
    v2f b1 = wrow[nc + 16];
    v2f b2 = wrow[nc + 32];
    v2f b3 = wrow[nc + 48];
    acc0 = __builtin_amdgcn_wmma_f32_16x16x4_f32(false, a, false, b0, (short)0, acc0, false, false);
    acc1 = __builtin_amdgcn_wmma_f32_16x16x4_f32(false, a, false, b1, (short)0, acc1, false, false);
    acc2 = __builtin_amdgcn_wmma_f32_16x16x4_f32(false, a, false, b2, (short)0, acc2, false, false);
    acc3 = __builtin_amdgcn_wmma_f32_16x16x4_f32(false, a, false, b3, (short)0, acc3, false, false);
  }

  int mr = (lane >> 4) << 3;          // C/D: lanes 16-31 hold M = 8 + vgpr
  for (int i = 0; i < 8; ++i) {
    float* yr = Y + (size_t)(r0 + mr + i) * 64 + nc;
    yr[0]  = acc0[i];
    yr[16] = acc1[i];
    yr[32] = acc2[i];
    yr[48] = acc3[i];
  }
}

// ---------------------------------------------------------------- WMMA  P += S^T * H
// P[64,64] += sum over a row-chunk of S^T H. 4 waves/block, wave w owns M-tile w.
__global__ __launch_bounds__(128) void ata64(const float* __restrict__ S,
                                             const float* __restrict__ H,
                                             float* __restrict__ P,
                                             int nrows, int chunk) {
  int tid = threadIdx.x;
  int wave = tid >> 5;
  int lane = tid & 31;
  int r0 = blockIdx.x * chunk;
  int r1 = min(r0 + chunk, nrows);    // chunk boundaries are multiples of 4
  int m     = lane & 15;
  int koff  = (lane >> 4) << 1;
  int mbase = wave * 16;

  v8f acc0 = {}, acc1 = {}, acc2 = {}, acc3 = {};
  for (int r = r0; r < r1; r += 4) {
    const float* s0 = S + (size_t)(r + koff) * 64 + mbase + m;   // A[m][k] = S[r+k][mbase+m]
    v2f a; a.x = s0[0]; a.y = s0[64];
    const float* h0 = H + (size_t)(r + koff) * 64 + (lane & 15); // B[k][n] = H[r+k][n]
    v2f b0; b0.x = h0[0];       b0.y = h0[64];
    v2f b1; b1.x = h0[16];      b1.y = h0[64 + 16];
    v2f b2; b2.x = h0[32];      b2.y = h0[64 + 32];
    v2f b3; b3.x = h0[48];      b3.y = h0[64 + 48];
    acc0 = __builtin_amdgcn_wmma_f32_16x16x4_f32(false, a, false, b0, (short)0, acc0, false, false);
    acc1 = __builtin_amdgcn_wmma_f32_16x16x4_f32(false, a, false, b1, (short)0, acc1, false, false);
    acc2 = __builtin_amdgcn_wmma_f32_16x16x4_f32(false, a, false, b2, (short)0, acc2, false, false);
    acc3 = __builtin_amdgcn_wmma_f32_16x16x4_f32(false, a, false, b3, (short)0, acc3, false, false);
  }

  int nc = lane & 15;
  int mr = (lane >> 4) << 3;
  for (int i = 0; i < 8; ++i) {
    float* pr = P + (size_t)(mbase + mr + i) * 64 + nc;
    atomicAdd(pr + 0,  acc0[i]);
    atomicAdd(pr + 16, acc1[i]);
    atomicAdd(pr + 32, acc2[i]);
    atomicAdd(pr + 48, acc3[i]);
  }
}

// ---------------------------------------------------------------- row softmax (64 wide)
__global__ void softmax64(float* __restrict__ S, int nrows) {
  long long t = (long long)blockIdx.x * blockDim.x + threadIdx.x;
  int r = (int)(t >> 5);
  if (r >= nrows) return;             // uniform per wave
  int lane = (int)t & 31;
  float2* p = (float2*)(S + (size_t)r * 64);
  float2 v = p[lane];
  float mx = fmaxf(v.x, v.y);
  for (int o = 16; o; o >>= 1) mx = fmaxf(mx, __shfl_xor(mx, o, 32));
  float e0 = __expf(v.x - mx);
  float e1 = __expf(v.y - mx);
  float sm = e0 + e1;
  for (int o = 16; o; o >>= 1) sm += __shfl_xor(sm, o, 32);
  float inv = 1.0f / sm;
  p[lane] = make_float2(e0 * inv, e1 * inv);
}

// ---------------------------------------------------------------- dense head (1 block)
__global__ __launch_bounds__(64) void final_head(const float* __restrict__ xp,
                                                 const float* __restrict__ adj,
                                                 const float* __restrict__ Wd,  const float* __restrict__ bd,
                                                 const float* __restrict__ Wl1, const float* __restrict__ bl1,
                                                 const float* __restrict__ Wl2, const float* __restrict__ bl2,
                                                 float* __restrict__ out) {
  __shared__ float s_xp[4096], s_adj[4096], s_t[4096], s_o[4096];
  __shared__ float d2[64], g[64], g1[64];
  int tid = threadIdx.x;
  for (int i = tid; i < 4096; i += 64) { s_xp[i] = xp[i]; s_adj[i] = adj[i]; }
  __syncthreads();
  s_adj[tid * 64 + tid] = 1.0f;       // set diagonal to 1
  __syncthreads();
  {
    float sum = 0.0f;
    for (int f = 0; f < 64; ++f) sum += s_adj[tid * 64 + f];
    d2[tid] = rsqrtf(fmaxf(sum, 1.0f));
  }
  __syncthreads();
  for (int c = 0; c < 64; ++c) {      // t = xp @ Wd  (thread owns column tid)
    float acc = 0.0f;
    for (int k = 0; k < 64; ++k) acc += s_xp[c * 64 + k] * Wd[k * 64 + tid];
    s_t[c * 64 + tid] = acc;
  }
  __syncthreads();
  float bdv = bd[tid];
  for (int c = 0; c < 64; ++c) {      // out = adjn @ t + bd
    float acc = 0.0f;
    float dc = d2[c];
    for (int k = 0; k < 64; ++k) acc += dc * s_adj[c * 64 + k] * d2[k] * s_t[k * 64 + tid];
    s_o[c * 64 + tid] = acc + bdv;
  }
  __syncthreads();
  {
    float acc = 0.0f;                 // mean over the 64 clusters
    for (int c = 0; c < 64; ++c) acc += s_o[c * 64 + tid];
    g[tid] = acc * (1.0f / 64.0f);
  }
  __syncthreads();
  {
    float acc = 0.0f;                 // relu(g @ Wl1 + bl1)
    for (int k = 0; k < 64; ++k) acc += g[k] * Wl1[k * 64 + tid];
    g1[tid] = fmaxf(acc + bl1[tid], 0.0f);
  }
  __syncthreads();
  if (tid == 0) {                     // g1 @ Wl2 + bl2, then log_softmax
    float g2[10];
    float mx = -1e30f;
    for (int j = 0; j < 10; ++j) {
      float acc = bl2[j];
      for (int k = 0; k < 64; ++k) acc += g1[k] * Wl2[k * 10 + j];
      g2[j] = acc;
      mx = fmaxf(mx, acc);
    }
    float sum = 0.0f;
    for (int j = 0; j < 10; ++j) sum += __expf(g2[j] - mx);
    float lse = mx + __logf(sum);
    for (int j = 0; j < 10; ++j) out[j] = g2[j] - lse;
  }
}

// ---------------------------------------------------------------- host orchestration
static inline long long cdiv_ll(long long a, long long b) { return (a + b - 1) / b; }

extern "C" void kernel_launch(void* const* d_in, const int* in_sizes, int n_in,
                              void* d_out, int out_size, void* d_ws, size_t ws_size,
                              hipStream_t stream) {
  const float* x   = (const float*)d_in[0];
  const float* W1a = (const float*)d_in[1];
  const float* b1a = (const float*)d_in[2];
  const float* W1b = (const float*)d_in[3];
  const float* b1b = (const float*)d_in[4];
  const float* Wpa = (const float*)d_in[5];
  const float* bpa = (const float*)d_in[6];
  const float* Wpb = (const float*)d_in[7];
  const float* bpb = (const float*)d_in[8];
  const float* Wd  = (const float*)d_in[9];
  const float* bd  = (const float*)d_in[10];
  const float* Wl1 = (const float*)d_in[11];
  const float* bl1 = (const float*)d_in[12];
  const float* Wl2 = (const float*)d_in[13];
  const float* bl2 = (const float*)d_in[14];
  const int*   edg = (const int*)d_in[15];

  const int N = in_sizes[0] / 64;
  const int E = in_sizes[15] / 2;
  const int* row0 = edg;       // edge_index[0]
  const int* col0 = edg + E;   // edge_index[1]

  const long long n64 = (long long)N * 64;

  // workspace layout
  float* buf0 = (float*)d_ws;
  float* buf1 = buf0 + n64;
  float* buf2 = buf1 + n64;
  float* dinv = buf2 + n64;
  float* xp   = dinv + N;
  float* adj  = xp + 4096;

  const int T = 256;
  dim3 blk(T);
  dim3 g_n(  (unsigned)cdiv_ll(N, T));
  dim3 g_e(  (unsigned)cdiv_ll(E, T));
  dim3 g_n64((unsigned)cdiv_ll(n64, T));
  dim3 g_e32((unsigned)cdiv_ll((long long)E * 32, T));
  dim3 g_sm( (unsigned)cdiv_ll((long long)N * 32, T));
  dim3 g_mm( (unsigned)cdiv_ll(cdiv_ll(N, 16), 4));
  const int CHUNK = 1024;
  dim3 g_ata((unsigned)cdiv_ll(N, CHUNK));

  // --- degrees -> dinv (self loop contributes the initial 1.0)
  fill_f32<<<g_n, blk, 0, stream>>>(dinv, 1.0f, N);
  deg_count<<<g_e, blk, 0, stream>>>(dinv, col0, E);
  deg_to_dinv<<<g_n, blk, 0, stream>>>(dinv, N);

  // --- GCN block 1: h = prop(relu(prop(x@W1a)+b1a) @ W1b) + b1b
  gemm64<<<g_mm, dim3(128), 0, stream>>>(x, W1a, buf0, N);
  prop_selfinit<<<g_n64, blk, 0, stream>>>(buf0, buf1, dinv, n64);
  edge_scatter<<<g_e32, blk, 0, stream>>>(buf0, buf1, row0, col0, dinv, E, 1);
  bias_act<<<g_n64, blk, 0, stream>>>(buf1, b1a, n64, 1);

  gemm64<<<g_mm, dim3(128), 0, stream>>>(buf1, W1b, buf0, N);
  prop_selfinit<<<g_n64, blk, 0, stream>>>(buf0, buf1, dinv, n64);
  edge_scatter<<<g_e32, blk, 0, stream>>>(buf0, buf1, row0, col0, dinv, E, 1);
  bias_act<<<g_n64, blk, 0, stream>>>(buf1, b1b, n64, 0);   // H in buf1

  // --- GCN block 2 -> s
  gemm64<<<g_mm, dim3(128), 0, stream>>>(buf1, Wpa, buf0, N);
  prop_selfinit<<<g_n64, blk, 0, stream>>>(buf0, buf2, dinv, n64);
  edge_scatter<<<g_e32, blk, 0, stream>>>(buf0, buf2, row0, col0, dinv, E, 1);
  bias_act<<<g_n64, blk, 0, stream>>>(buf2, bpa, n64, 1);

  gemm64<<<g_mm, dim3(128), 0, stream>>>(buf2, Wpb, buf0, N);
  prop_selfinit<<<g_n64, blk, 0, stream>>>(buf0, buf2, dinv, n64);
  edge_scatter<<<g_e32, blk, 0, stream>>>(buf0, buf2, row0, col0, dinv, E, 1);
  bias_act<<<g_n64, blk, 0, stream>>>(buf2, bpb, n64, 0);

  softmax64<<<g_sm, blk, 0, stream>>>(buf2, N);             // S in buf2

  // --- As[row0] += S[col0]  (buf0)
  fill_f32<<<g_n64, blk, 0, stream>>>(buf0, 0.0f, n64);
  edge_scatter<<<g_e32, blk, 0, stream>>>(buf2, buf0, col0, row0, dinv, E, 0);

  // --- xp = S^T H ; adj = S^T As
  fill_f32<<<dim3(32), blk, 0, stream>>>(xp, 0.0f, 8192);   // zero xp and adj (contiguous)
  ata64<<<g_ata, dim3(128), 0, stream>>>(buf2, buf1, xp,  N, CHUNK);
  ata64<<<g_ata, dim3(128), 0, stream>>>(buf2, buf0, adj, N, CHUNK);

  // --- dense diff-pool head + MLP + log_softmax
  final_head<<<dim3(1), dim3(64), 0, stream>>>(xp, adj, Wd, bd, Wl1, bl1, Wl2, bl2,
                                               (float*)d_out);
}